// STABLESSMKernel_75565654606232
// MI455X (gfx1250) — compile-verified
//
#include <hip/hip_runtime.h>
#include <hip/hip_bf16.h>

// S4D / Vandermonde SSM kernel for MI455X (gfx1250, wave32).
//
// K[h, l] = 2 * Re( sum_n Ct[h,n] * w[h,n]^l ),  w = exp(dt*A)
// Split l = 64*l1 + l0:  K[h, 64*l1 + l0] = sum_k A_h[l1,k] * B_h[k,l0]
//   A_h[l1, n]    =  2*Re(Ct * w^(64*l1))     (k = n      , n = 0..31)
//   A_h[l1, 32+n] = -2*Im(Ct * w^(64*l1))     (k = 32 + n)
//   B_h[n, l0]    =  Re(w^l0),  B_h[32+n, l0] = Im(w^l0)
// -> per-channel real GEMM 64x64x64, done with V_WMMA_F32_16X16X4_F32.

typedef float v2f __attribute__((ext_vector_type(2)));
typedef float v8f __attribute__((ext_vector_type(8)));

#define H_DIM   1024
#define NHALF   32      // complex modes per channel
#define L_TOT   4096
#define BSUB    64      // l0 block; L_TOT / BSUB = 64 rows (l1)
#define APITCH  65      // column-access friendly (bank-conflict free down a column)
#define BPITCH  72      // row-access friendly; rows k and k+2 land in disjoint banks

__global__ __launch_bounds__(256)
void ssm_vandermonde_wmma(const float* __restrict__ C_real,
                          const float* __restrict__ log_dt,
                          const float* __restrict__ log_A_real,
                          const float* __restrict__ A_imag,
                          float* __restrict__ out)
{
    __shared__ float sA[BSUB * APITCH];            // P tile: (l1) x (k)
    __shared__ float sB[BSUB * BPITCH];            // Q tile: (k)  x (l0)
    __shared__ float sCtr[NHALF], sCti[NHALF];
    __shared__ float sDr[NHALF],  sDi[NHALF];      // dtA (real, imag)

    const int h   = blockIdx.x;
    const int tid = threadIdx.x;

    // ---- per-channel parameters: dt, A, dtA, Ct = C*(exp(dtA)-1)/A ----
    if (tid < NHALF) {
        const int n = tid;
        const float dt   = __expf(log_dt[h]);
        const float Ar   = -__expf(log_A_real[h * NHALF + n]);
        const float Ai   = A_imag[h * NHALF + n];
        const float dtAr = Ar * dt;
        const float dtAi = Ai * dt;
        const float Cr   = C_real[(h * NHALF + n) * 2 + 0];
        const float Ci   = C_real[(h * NHALF + n) * 2 + 1];

        float s, c;
        const float er = __expf(dtAr);
        __sincosf(dtAi, &s, &c);
        const float Er  = er * c - 1.0f;           // exp(dtA) - 1
        const float Ei  = er * s;
        const float inv = 1.0f / (Ar * Ar + Ai * Ai);
        const float Dr  = (Er * Ar + Ei * Ai) * inv;   // (exp(dtA)-1)/A
        const float Di  = (Ei * Ar - Er * Ai) * inv;

        sCtr[n] = Cr * Dr - Ci * Di;
        sCti[n] = Cr * Di + Ci * Dr;
        sDr[n]  = dtAr;
        sDi[n]  = dtAi;
    }
    __syncthreads();

    // ---- B tile (Q): w^l0 for l0 = 0..63 ----
    for (int i = tid; i < NHALF * BSUB; i += 256) {
        const int n  = i >> 6;
        const int l0 = i & 63;
        const float x = (float)l0;
        float s, c;
        const float r = __expf(sDr[n] * x);
        __sincosf(sDi[n] * x, &s, &c);
        sB[n * BPITCH + l0]           = r * c;     // Re
        sB[(NHALF + n) * BPITCH + l0] = r * s;     // Im
    }

    // ---- A tile (P): Ct * w^(64*l1) for l1 = 0..63 (scale 2 / -2 folded in) ----
    for (int i = tid; i < NHALF * BSUB; i += 256) {
        const int n  = i & 31;
        const int l1 = i >> 5;
        const float x = (float)(BSUB * l1);
        float s, c;
        const float r = __expf(sDr[n] * x);
        __sincosf(sDi[n] * x, &s, &c);
        const float pr = r * (sCtr[n] * c - sCti[n] * s);
        const float pi = r * (sCtr[n] * s + sCti[n] * c);
        sA[l1 * APITCH + n]         =  2.0f * pr;
        sA[l1 * APITCH + NHALF + n] = -2.0f * pi;
    }
    __syncthreads();

    // ---- 64x64x64 GEMM via V_WMMA_F32_16X16X4_F32 ----
    // 4x4 grid of 16x16 output tiles; 8 waves, 2 tiles per wave.
    const int wave = tid >> 5;
    const int lane = tid & 31;
    const int mloc = lane & 15;     // M (A rows) or N (B cols) within tile
    const int ksel = lane >> 4;     // lanes 16-31 hold K+2/K+3 (A) and M+8 (C/D)

    for (int t = 2 * wave; t < 2 * wave + 2; ++t) {
        const int ti = t >> 2;      // l1 tile
        const int tj = t & 3;       // l0 tile
        v8f acc = {};
        #pragma unroll
        for (int kk = 0; kk < 16; ++kk) {
            const int k0 = kk * 4 + 2 * ksel;
            v2f a, b;
            // 32-bit A 16x4 layout: lanes 0-15 -> K={k,k+1}, lanes 16-31 -> K={k+2,k+3}
            a.x = sA[(ti * 16 + mloc) * APITCH + k0];
            a.y = sA[(ti * 16 + mloc) * APITCH + k0 + 1];
            // B 4x16: rows striped across lanes within a VGPR, mirrored K split
            b.x = sB[(k0)     * BPITCH + tj * 16 + mloc];
            b.y = sB[(k0 + 1) * BPITCH + tj * 16 + mloc];
            acc = __builtin_amdgcn_wmma_f32_16x16x4_f32(
                      /*neg_a=*/false, a, /*neg_b=*/false, b,
                      /*c_mod=*/(short)0, acc,
                      /*reuse_a=*/false, /*reuse_b=*/false);
        }
        // C/D 16x16 f32 layout: VGPR r -> M = r (lanes 0-15) / r+8 (lanes 16-31)
        #pragma unroll
        for (int r = 0; r < 8; ++r) {
            const int l1 = ti * 16 + r + 8 * ksel;
            const int l0 = tj * 16 + mloc;
            out[h * L_TOT + l1 * BSUB + l0] = acc[r];
        }
    }
}

extern "C" void kernel_launch(void* const* d_in, const int* in_sizes, int n_in,
                              void* d_out, int out_size, void* d_ws, size_t ws_size,
                              hipStream_t stream) {
    (void)in_sizes; (void)n_in; (void)out_size; (void)d_ws; (void)ws_size;
    const float* C_real     = (const float*)d_in[0];   // (H, 32, 2)
    const float* log_dt     = (const float*)d_in[1];   // (H,)
    const float* log_A_real = (const float*)d_in[2];   // (H, 32)
    const float* A_imag     = (const float*)d_in[3];   // (H, 32)
    float* out = (float*)d_out;                        // (H, L)

    ssm_vandermonde_wmma<<<H_DIM, 256, 0, stream>>>(
        C_real, log_dt, log_A_real, A_imag, out);
}